// SSIMAELoss_61838939127961
// MI455X (gfx1250) — compile-verified
//
#include <hip/hip_runtime.h>
#include <hip/hip_bf16.h>

// ---------------- problem constants ----------------
#define BATCH 32
#define H0 480
#define W0 640
#define N0 (H0 * W0)            // 307200
#define RANK 153599u            // (N0-1)/2, 0-based median index
#define NBLK 64                 // reduction blocks per image
#define NBLK_SSI 128
#define HBINS 65536             // 16-bit radix bins

typedef __attribute__((ext_vector_type(2))) float v2f;
typedef __attribute__((ext_vector_type(8))) float v8f;
typedef unsigned int v4u __attribute__((ext_vector_type(4)));
typedef int v4i __attribute__((ext_vector_type(4)));
typedef int v8i __attribute__((ext_vector_type(8)));

#if defined(__HIP_DEVICE_COMPILE__) && __has_builtin(__builtin_amdgcn_tensor_load_to_lds) && __has_builtin(__builtin_amdgcn_s_wait_tensorcnt)
#define USE_TDM 1
#else
#define USE_TDM 0
#endif

// ---------------- float <-> order-preserving key ----------------
__device__ __forceinline__ unsigned fkey(float f) {
    unsigned u = __float_as_uint(f);
    return u ^ ((unsigned)((int)u >> 31) | 0x80000000u);
}
__device__ __forceinline__ float fromkey(unsigned k) {
    unsigned u = k ^ ((k & 0x80000000u) ? 0x80000000u : 0xFFFFFFFFu);
    return __uint_as_float(u);
}

// ---------------- init state ----------------
__global__ void k_init(unsigned* prefix, unsigned* remaining, unsigned* hist,
                       float* sabs, float* gl) {
    int i = blockIdx.x * blockDim.x + threadIdx.x;
    if (i < 2 * BATCH) { prefix[i] = 0u; remaining[i] = RANK; }
    if (i < BATCH) { sabs[i] = 0.0f; gl[i] = 0.0f; }
    int stride = gridDim.x * blockDim.x;
    for (int h = i; h < 2 * BATCH * HBINS; h += stride) hist[h] = 0u;
}

// ---------------- 16-bit radix select: histogram pass ----------------
__global__ void k_hist16(const float* __restrict__ pred, const float* __restrict__ tgt,
                         unsigned* hist, const unsigned* prefix, int pass) {
    int img = blockIdx.y, t = blockIdx.z, id = t * BATCH + img;
    const float* src = (t == 0 ? pred : tgt) + (size_t)img * N0;
    unsigned* h = hist + (size_t)id * HBINS;
    unsigned pfxHigh = prefix[id] >> 16;
    int stride = gridDim.x * blockDim.x;
    const float4* s4 = (const float4*)src;
    for (int i = blockIdx.x * blockDim.x + threadIdx.x; i < N0 / 4; i += stride) {
        float4 x = s4[i];
        unsigned k0 = fkey(x.x), k1 = fkey(x.y), k2 = fkey(x.z), k3 = fkey(x.w);
        if (pass == 0) {
            atomicAdd(&h[k0 >> 16], 1u);
            atomicAdd(&h[k1 >> 16], 1u);
            atomicAdd(&h[k2 >> 16], 1u);
            atomicAdd(&h[k3 >> 16], 1u);
        } else {
            if ((k0 >> 16) == pfxHigh) atomicAdd(&h[k0 & 0xFFFFu], 1u);
            if ((k1 >> 16) == pfxHigh) atomicAdd(&h[k1 & 0xFFFFu], 1u);
            if ((k2 >> 16) == pfxHigh) atomicAdd(&h[k2 & 0xFFFFu], 1u);
            if ((k3 >> 16) == pfxHigh) atomicAdd(&h[k3 & 0xFFFFu], 1u);
        }
    }
}

// ---------------- 16-bit radix select: hierarchical pick ----------------
__global__ void k_select16(unsigned* hist, unsigned* prefix, unsigned* remaining,
                           float* shiftv, int pass) {
    int img = blockIdx.x, t = blockIdx.y, id = t * BATCH + img;
    unsigned* h = hist + (size_t)id * HBINS;
    int tid = threadIdx.x;
    __shared__ unsigned csum[256], scan[256], binbuf[256];
    __shared__ int selChunk;
    __shared__ unsigned selRem;
    // each thread sums its 256-bin chunk
    unsigned s = 0;
    int base = tid * 256;
    for (int j = 0; j < 256; ++j) s += h[base + j];
    csum[tid] = s;
    scan[tid] = s;
    __syncthreads();
    // inclusive scan over chunk sums (Hillis-Steele)
    for (int o = 1; o < 256; o <<= 1) {
        unsigned add = (tid >= o) ? scan[tid - o] : 0u;
        __syncthreads();
        scan[tid] += add;
        __syncthreads();
    }
    if (tid == 0) {
        unsigned r = remaining[id];
        int c = 255;
        for (int i = 0; i < 256; ++i) {
            if (r < scan[i]) { c = i; break; }
        }
        selChunk = c;
        selRem = r - (scan[c] - csum[c]);   // subtract exclusive prefix
    }
    __syncthreads();
    binbuf[tid] = h[selChunk * 256 + tid];  // stage winning chunk into LDS
    __syncthreads();
    if (tid == 0) {
        unsigned r = selRem;
        int b = 255;
        for (int j = 0; j < 256; ++j) {
            if (r < binbuf[j]) { b = j; break; }
            r -= binbuf[j];
        }
        unsigned val16 = (unsigned)(selChunk * 256 + b);
        unsigned pfx = prefix[id];
        pfx = (pass == 0) ? (val16 << 16) : (pfx | val16);
        prefix[id] = pfx;
        remaining[id] = r;
        if (pass == 1) shiftv[id] = fromkey(pfx);
    }
    __syncthreads();
    // re-zero this image's histogram for the next pass
    for (int j = tid; j < HBINS; j += 256) h[j] = 0u;
}

// ---------------- |x - shift| partial sums (for scale) ----------------
__global__ void k_abspart(const float* __restrict__ pred, const float* __restrict__ tgt,
                          const float* shiftv, float* partial) {
    int img = blockIdx.y, t = blockIdx.z;
    const float* src = (t == 0 ? pred : tgt) + (size_t)img * N0;
    float sh = shiftv[t * BATCH + img];
    float s = 0.0f;
    int stride = gridDim.x * blockDim.x;
    const float4* s4 = (const float4*)src;
    for (int i = blockIdx.x * blockDim.x + threadIdx.x; i < N0 / 4; i += stride) {
        float4 x = s4[i];
        s += fabsf(x.x - sh) + fabsf(x.y - sh) + fabsf(x.z - sh) + fabsf(x.w - sh);
    }
    __shared__ float red[256];
    red[threadIdx.x] = s;
    __syncthreads();
    for (int o = 128; o > 0; o >>= 1) {
        if (threadIdx.x < o) red[threadIdx.x] += red[threadIdx.x + o];
        __syncthreads();
    }
    if (threadIdx.x == 0) partial[(t * BATCH + img) * NBLK_SSI + blockIdx.x] = red[0];
}

__global__ void k_scale_fin(const float* partial, float* scalev) {
    int img = blockIdx.x, t = blockIdx.y, id = t * BATCH + img;
    __shared__ float red[NBLK];
    red[threadIdx.x] = partial[id * NBLK_SSI + threadIdx.x];
    __syncthreads();
    for (int o = NBLK / 2; o > 0; o >>= 1) {
        if (threadIdx.x < o) red[threadIdx.x] += red[threadIdx.x + o];
        __syncthreads();
    }
    if (threadIdx.x == 0) scalev[id] = red[0] * (1.0f / (float)N0);
}

// ---------------- fused SSI normalize + |p - t| partials ----------------
__global__ void k_ssi(const float* __restrict__ pred, const float* __restrict__ tgt,
                      const float* shiftv, const float* scalev,
                      float* __restrict__ p0, float* __restrict__ t0, float* partial) {
    int img = blockIdx.y;
    size_t base4 = (size_t)img * (N0 / 4);
    float shp = shiftv[img], sht = shiftv[BATCH + img];
    float rsp = 1.0f / scalev[img], rst = 1.0f / scalev[BATCH + img];
    float s = 0.0f;
    int stride = gridDim.x * blockDim.x;
    const float4* pr4 = (const float4*)pred;
    const float4* tg4 = (const float4*)tgt;
    float4* p4 = (float4*)p0;
    float4* t4 = (float4*)t0;
    for (int i = blockIdx.x * blockDim.x + threadIdx.x; i < N0 / 4; i += stride) {
        __builtin_prefetch(&pr4[base4 + i + stride], 0, 0);   // global_prefetch_b8
        __builtin_prefetch(&tg4[base4 + i + stride], 0, 0);
        float4 xp = pr4[base4 + i];
        float4 xt = tg4[base4 + i];
        float4 a, b;
        a.x = (xp.x - shp) * rsp; a.y = (xp.y - shp) * rsp;
        a.z = (xp.z - shp) * rsp; a.w = (xp.w - shp) * rsp;
        b.x = (xt.x - sht) * rst; b.y = (xt.y - sht) * rst;
        b.z = (xt.z - sht) * rst; b.w = (xt.w - sht) * rst;
        p4[base4 + i] = a;
        t4[base4 + i] = b;
        s += fabsf(a.x - b.x) + fabsf(a.y - b.y) + fabsf(a.z - b.z) + fabsf(a.w - b.w);
    }
    __shared__ float red[256];
    red[threadIdx.x] = s;
    __syncthreads();
    for (int o = 128; o > 0; o >>= 1) {
        if (threadIdx.x < o) red[threadIdx.x] += red[threadIdx.x + o];
        __syncthreads();
    }
    if (threadIdx.x == 0) partial[img * NBLK_SSI + blockIdx.x] = red[0];
}

__global__ void k_sabs_fin(const float* partial, float* sabs) {
    int img = blockIdx.x;
    __shared__ float red[NBLK_SSI];
    red[threadIdx.x] = partial[img * NBLK_SSI + threadIdx.x];
    __syncthreads();
    for (int o = NBLK_SSI / 2; o > 0; o >>= 1) {
        if (threadIdx.x < o) red[threadIdx.x] += red[threadIdx.x + o];
        __syncthreads();
    }
    if (threadIdx.x == 0) sabs[img] = red[0] * (1.0f / (float)N0);  // mean |p-t|
}

// ---------------- Scharr gradient magnitude ----------------
__device__ __forceinline__ float gradmag(const float* __restrict__ x, int r, int c, int W) {
    float a = x[(r - 1) * W + c - 1], b = x[(r - 1) * W + c], cc = x[(r - 1) * W + c + 1];
    float d = x[r * W + c - 1],                                e = x[r * W + c + 1];
    float f = x[(r + 1) * W + c - 1], g = x[(r + 1) * W + c], h = x[(r + 1) * W + c + 1];
    float gx = 3.0f * (a - cc) + 10.0f * (d - e) + 3.0f * (f - h);
    float gy = 3.0f * (a - f) + 10.0f * (b - g) + 3.0f * (cc - h);
    return sqrtf(gx * gx + gy * gy);
}
__device__ __forceinline__ float gradval(const float* __restrict__ p, const float* __restrict__ t,
                                         int i, int Wi, int W, int M) {
    if (i >= M) return 0.0f;
    int r = i / Wi + 1, c = i % Wi + 1;
    return fabsf(gradmag(p, r, c, W) - gradmag(t, r, c, W));
}

// ---------------- gradient loss with WMMA reduction ----------------
// A = ones(16x4) => D(m,n) = sum_k B(k,n) + C(m,n); every row of D replicates
// the column-sums of B, so Sum(D) == 16 * Sum(all B fed in). Exact in f32 and
// independent of the B VGPR striping. 64 values reduced per v_wmma issue.
__global__ void k_grad(const float* __restrict__ P, const float* __restrict__ T,
                       int H, int W, float* partial) {
    int img = blockIdx.y;
    const float* p = P + (size_t)img * H * W;
    const float* t = T + (size_t)img * H * W;
    int Wi = W - 2;
    int M = (H - 2) * Wi;
    v2f ones = {1.0f, 1.0f};
    v8f acc = {};
    int stride = gridDim.x * blockDim.x * 2;
    // block-uniform loop bound: every lane executes every WMMA (EXEC all-1s)
    for (int base = blockIdx.x * blockDim.x * 2; base < M; base += stride) {
        int i0 = base + threadIdx.x * 2;
        float v0 = gradval(p, t, i0, Wi, W, M);
        float v1 = gradval(p, t, i0 + 1, Wi, W, M);
        v2f bv = {v0, v1};
        acc = __builtin_amdgcn_wmma_f32_16x16x4_f32(
            /*neg_a=*/false, ones, /*neg_b=*/false, bv,
            /*c_mod=*/(short)0, acc, /*reuse_a=*/false, /*reuse_b=*/false);
    }
    float s = acc[0] + acc[1] + acc[2] + acc[3] + acc[4] + acc[5] + acc[6] + acc[7];
    s += __shfl_down(s, 16);
    s += __shfl_down(s, 8);
    s += __shfl_down(s, 4);
    s += __shfl_down(s, 2);
    s += __shfl_down(s, 1);
    __shared__ float wsum[8];
    int wave = threadIdx.x >> 5, lane = threadIdx.x & 31;
    if (lane == 0) wsum[wave] = s;
    __syncthreads();
    if (threadIdx.x == 0) {
        float tot = 0.0f;
        for (int w = 0; w < 8; ++w) tot += wsum[w];
        partial[img * NBLK_SSI + blockIdx.x] = tot * (1.0f / 16.0f);  // undo 16x replication
    }
}

__global__ void k_grad_fin(const float* partial, float* gl, float invM) {
    int img = blockIdx.x;
    __shared__ float red[NBLK];
    red[threadIdx.x] = partial[img * NBLK_SSI + threadIdx.x];
    __syncthreads();
    for (int o = NBLK / 2; o > 0; o >>= 1) {
        if (threadIdx.x < o) red[threadIdx.x] += red[threadIdx.x + o];
        __syncthreads();
    }
    if (threadIdx.x == 0) gl[img] += red[0] * invM;
}

// ---------------- TDM: load nrows x W f32 rows into LDS ----------------
#if USE_TDM
__device__ __forceinline__ void tdm_load_rows(const float* gsrc, unsigned ldsOff,
                                              int nrows, int W, int Hdim) {
    unsigned long long ga = (unsigned long long)(const void*)gsrc;
    v4u g0;
    g0.x = 1u;                                            // count=1 (valid), user mode
    g0.y = ldsOff;                                        // lds_addr (bytes)
    g0.z = (unsigned)(ga & 0xFFFFFFFFull);                // global_addr[31:0]
    g0.w = (unsigned)((ga >> 32) & 0x1FFFFFFull) | (2u << 30);  // [56:32] | type=2
    v8i g1;
    g1[0] = (int)(2u << 16);                              // data_size=2 (4 bytes)
    g1[1] = (int)(((unsigned)W & 0xFFFFu) << 16);         // tensor_dim0[15:0]
    g1[2] = (int)((((unsigned)W >> 16) & 0xFFFFu) |
                  (((unsigned)Hdim & 0xFFFFu) << 16));    // dim0 hi | tensor_dim1 lo
    g1[3] = (int)((((unsigned)Hdim >> 16) & 0xFFFFu) |
                  (((unsigned)W & 0xFFFFu) << 16));       // dim1 hi | tile_dim0 = W
    g1[4] = (int)((unsigned)nrows & 0xFFFFu);             // tile_dim1 = nrows
    g1[5] = (int)(unsigned)W;                             // tensor_dim0_stride = W (lo32)
    g1[6] = 0;
    g1[7] = 0;
    v4i gz = {0, 0, 0, 0};
#if __clang_major__ >= 23
    v8i gz8 = {0, 0, 0, 0, 0, 0, 0, 0};
    __builtin_amdgcn_tensor_load_to_lds(g0, g1, gz, gz, gz8, 0);
#else
    __builtin_amdgcn_tensor_load_to_lds(g0, g1, gz, gz, 0);
#endif
}
#endif

// ---------------- gauss(5x5, pad 2) + 2x2 avgpool == separable 6x6 ----------------
// 1D combined filter: [1,5,10,10,5,1]/32 (zero-padded input), stride 2.
// One output row per block; 6-row input band staged in LDS (TDM when available).
__global__ void k_down(const float* __restrict__ sp, const float* __restrict__ st,
                       float* __restrict__ dp, float* __restrict__ dt, int H, int W) {
    int Ho = H / 2, Wo = W / 2;
    int img = blockIdx.y;
    const float* s = (blockIdx.z ? st : sp) + (size_t)img * H * W;
    float* d = (blockIdx.z ? dt : dp) + (size_t)img * Ho * Wo;
    int oi = blockIdx.x;
    __shared__ float tile[6 * W0];
    int r0 = 2 * oi - 2;
    int rlo = r0 < 0 ? 0 : r0;
    int rhi = (r0 + 5) < (H - 1) ? (r0 + 5) : (H - 1);
    int z0 = rlo - r0;            // leading zero rows
    int nr = rhi - rlo + 1;       // valid rows
    for (int j = threadIdx.x; j < z0 * W; j += (int)blockDim.x) tile[j] = 0.0f;
    for (int j = threadIdx.x; j < (6 - z0 - nr) * W; j += (int)blockDim.x)
        tile[(z0 + nr) * W + j] = 0.0f;
#if USE_TDM
    if (threadIdx.x < 32) {       // one wave issues the DMA
        tdm_load_rows(s + (size_t)rlo * W,
                      (unsigned)(unsigned long long)(void*)&tile[z0 * W], nr, W, H);
        __builtin_amdgcn_s_wait_tensorcnt(0);
    }
    __syncthreads();
#else
    for (int j = threadIdx.x; j < nr * W; j += (int)blockDim.x)
        tile[z0 * W + j] = s[(size_t)rlo * W + j];
    __syncthreads();
#endif
    const float cf[6] = {1.0f / 32, 5.0f / 32, 10.0f / 32, 10.0f / 32, 5.0f / 32, 1.0f / 32};
    for (int oj = threadIdx.x; oj < Wo; oj += (int)blockDim.x) {
        int c0 = 2 * oj - 2;
        float acc = 0.0f;
        #pragma unroll
        for (int a = 0; a < 6; ++a) {
            float rowsum = 0.0f;
            #pragma unroll
            for (int b = 0; b < 6; ++b) {
                int c = c0 + b;
                if (c >= 0 && c < W) rowsum += cf[b] * tile[a * W + c];
            }
            acc += cf[a] * rowsum;
        }
        d[(size_t)oi * Wo + oj] = acc;
    }
}

// ---------------- final combine ----------------
__global__ void k_final(const float* sabs, const float* gl, float* out) {
    __shared__ float red[BATCH];
    float v = sabs[threadIdx.x] * 0.5f + 0.5f * (gl[threadIdx.x] * 0.25f);
    red[threadIdx.x] = v;
    __syncthreads();
    for (int o = BATCH / 2; o > 0; o >>= 1) {
        if (threadIdx.x < o) red[threadIdx.x] += red[threadIdx.x + o];
        __syncthreads();
    }
    if (threadIdx.x == 0) out[0] = red[0] * (1.0f / (float)BATCH);
}

// ---------------- host launcher ----------------
extern "C" void kernel_launch(void* const* d_in, const int* in_sizes, int n_in,
                              void* d_out, int out_size, void* d_ws, size_t ws_size,
                              hipStream_t stream) {
    const float* pred = (const float*)d_in[0];
    const float* tgt  = (const float*)d_in[1];
    float* out = (float*)d_out;

    char* ws = (char*)d_ws;
    unsigned* prefix    = (unsigned*)(ws + 0);          // 64 u32
    unsigned* remaining = (unsigned*)(ws + 256);        // 64 u32
    float*    shiftv    = (float*)(ws + 512);           // 64 f
    float*    scalev    = (float*)(ws + 768);           // 64 f
    float*    sabs      = (float*)(ws + 1024);          // 32 f
    float*    gl        = (float*)(ws + 1280);          // 32 f
    float*    partial   = (float*)(ws + 1536);          // 2*32*128 f = 32 KB
    unsigned* hist      = (unsigned*)(ws + 34304);      // 2*32*65536 u32 = 16 MB
    size_t off = 34304 + (size_t)2 * BATCH * HBINS * 4; // = 16,811,520
    const int Hs[4]  = {480, 240, 120, 60};
    const int Ws4[4] = {640, 320, 160, 80};
    float* pl[4]; float* tl[4];
    for (int s = 0; s < 4; ++s) {
        size_t n = (size_t)BATCH * Hs[s] * Ws4[s] * sizeof(float);
        pl[s] = (float*)(ws + off); off += n;
        tl[s] = (float*)(ws + off); off += n;
    }
    (void)ws_size; (void)in_sizes; (void)n_in; (void)out_size;

    k_init<<<2048, 256, 0, stream>>>(prefix, remaining, hist, sabs, gl);

    // 2-pass 16-bit radix select: per-image median of the float-ordered key
    for (int pass = 0; pass < 2; ++pass) {
        k_hist16<<<dim3(NBLK, BATCH, 2), 256, 0, stream>>>(pred, tgt, hist, prefix, pass);
        k_select16<<<dim3(BATCH, 2), 256, 0, stream>>>(hist, prefix, remaining, shiftv, pass);
    }

    // scale = mean |x - median|
    k_abspart<<<dim3(NBLK, BATCH, 2), 256, 0, stream>>>(pred, tgt, shiftv, partial);
    k_scale_fin<<<dim3(BATCH, 2), NBLK, 0, stream>>>(partial, scalev);

    // SSI normalize + L1 term
    k_ssi<<<dim3(NBLK_SSI, BATCH), 256, 0, stream>>>(pred, tgt, shiftv, scalev,
                                                     pl[0], tl[0], partial);
    k_sabs_fin<<<BATCH, NBLK_SSI, 0, stream>>>(partial, sabs);

    // multi-scale gradient loss
    for (int s = 0; s < 4; ++s) {
        int H = Hs[s], W = Ws4[s];
        float invM = 1.0f / (float)((H - 2) * (W - 2));
        k_grad<<<dim3(NBLK, BATCH), 256, 0, stream>>>(pl[s], tl[s], H, W, partial);
        k_grad_fin<<<BATCH, NBLK, 0, stream>>>(partial, gl, invM);
        if (s < 3)
            k_down<<<dim3(Hs[s + 1], BATCH, 2), 256, 0, stream>>>(pl[s], tl[s],
                                                                  pl[s + 1], tl[s + 1], H, W);
    }

    k_final<<<1, BATCH, 0, stream>>>(sabs, gl, out);
}